// GMNet_51677046505960
// MI455X (gfx1250) — compile-verified
//
#include <hip/hip_runtime.h>
#include <hip/hip_bf16.h>
#include <math.h>

// ---------------------------------------------------------------------------
// GMNet forward for MI455X (gfx1250, wave32).
//   Phase A: f32 WMMA GEMMs (v_wmma_f32_16x16x4_f32), templated per transpose
//            mode so each inner loop is branch-free.
//   Phase B: single persistent workgroup runs the 150-iter ADMM QP using the
//            Kronecker/low-rank structure of Q and C (O(n1*n2) per matvec,
//            warm-started CG inner solve), then fused clip+softmax.
// ---------------------------------------------------------------------------

typedef __attribute__((ext_vector_type(2))) float v2f;
typedef __attribute__((ext_vector_type(8))) float v8f;

#define ALPHA_F   200.0f
#define RHO_F     100.0f
#define SIGMA_F   1e-3f
#define ADMM_ITERS 150
#define CG_ITERS   30
#define KDIAG     (3969.0f + SIGMA_F + RHO_F)   // (n1-1)(n2-1) + sigma + rho

// ---------------------------------------------------------------------------
// WMMA GEMM: D(MxN) = op(A) * op(B) [+Cin] [+bias,relu]
//   TA=0: A is MxK row-major (lda)   TA=1: A is KxM row-major (lda)
//   TB=1: B is NxK row-major (ldb)   TB=0: B is KxN row-major (ldb)
// One wave per 16x16 tile; K stepped by 4 per v_wmma_f32_16x16x4_f32,
// unrolled x4. f32 A-frag layout (ISA 7.12.2): lanes 0-15 K={0,1},
// lanes 16-31 K={2,3}. C/D: VGPR r -> M=r (lo lanes) / r+8 (hi), N=lane&15.
// ---------------------------------------------------------------------------
template <int TA, int TB, bool HASC, bool BIASRELU>
__global__ __launch_bounds__(256) void k_wmma_gemm(
    const float* __restrict__ A, const float* __restrict__ B,
    const float* __restrict__ Cin, const float* __restrict__ bias,
    float* __restrict__ D, int M, int N, int K,
    int lda, int ldb, int ldd)
{
  const int wave   = (int)((blockIdx.x * blockDim.x + threadIdx.x) >> 5);
  const int lane   = (int)(threadIdx.x & 31);
  const int tilesN = N >> 4;
  const int tiles  = (M >> 4) * tilesN;
  if (wave >= tiles) return;                 // whole-wave uniform exit
  const int tm   = (wave / tilesN) << 4;
  const int tn   = (wave % tilesN) << 4;
  const int half = lane >> 4;                // 0: lanes 0-15, 1: lanes 16-31
  const int l15  = lane & 15;

  v8f acc = {};
  if constexpr (HASC) {
#pragma unroll
    for (int rr = 0; rr < 8; ++rr)
      acc[rr] = Cin[(tm + rr + (half << 3)) * ldd + tn + l15];
  }

  const float* arow = A + (tm + l15) * lda;  // used when TA==0 (contiguous K)
  const float* brow = B + (tn + l15) * ldb;  // used when TB==1 (contiguous K)
  const float* acol = A + (tm + l15);        // used when TA==1 (K along rows)
  const float* bcol = B + (tn + l15);        // used when TB==0

  for (int k0 = 0; k0 < K; k0 += 16) {
    // prefetch one unroll-group ahead on the streamed operands
    if (k0 + 16 < K) {
      if constexpr (TA == 0) __builtin_prefetch(arow + k0 + 16, 0, 1);
      if constexpr (TB == 1) __builtin_prefetch(brow + k0 + 16, 0, 1);
    }
#pragma unroll
    for (int u = 0; u < 4; ++u) {
      const int kk = k0 + (u << 2) + (half << 1);
      v2f av, bv;
      if constexpr (TA) { av.x = acol[kk * lda]; av.y = acol[(kk + 1) * lda]; }
      else              { av = *(const v2f*)(arow + kk); }
      if constexpr (TB) { bv = *(const v2f*)(brow + kk); }
      else              { bv.x = bcol[kk * ldb]; bv.y = bcol[(kk + 1) * ldb]; }
      acc = __builtin_amdgcn_wmma_f32_16x16x4_f32(
          /*neg_a=*/false, av, /*neg_b=*/false, bv,
          /*c_mod=*/(short)0, acc, /*reuse_a=*/false, /*reuse_b=*/false);
    }
  }

  const int col = tn + l15;
#pragma unroll
  for (int rr = 0; rr < 8; ++rr) {
    float v = acc[rr];
    if constexpr (BIASRELU) v = fmaxf(v + bias[col], 0.f);
    D[(tm + rr + (half << 3)) * ldd + col] = v;
  }
}

// ---------------------------------------------------------------------------
// u = f + (||f||/||m||) * m   (one block per row; deterministic LDS tree)
// ---------------------------------------------------------------------------
__global__ __launch_bounds__(256) void k_fuse_lam(
    const float* __restrict__ F, const float* __restrict__ Mm,
    float* __restrict__ U, int d)
{
  __shared__ float s1[256], s2[256];
  const int row = blockIdx.x, t = threadIdx.x;
  float nf = 0.f, nm = 0.f;
  for (int c = t; c < d; c += 256) {
    const float f = F[row * d + c], m = Mm[row * d + c];
    nf += f * f; nm += m * m;
  }
  s1[t] = nf; s2[t] = nm; __syncthreads();
  for (int s = 128; s > 0; s >>= 1) {
    if (t < s) { s1[t] += s1[t + s]; s2[t] += s2[t + s]; }
    __syncthreads();
  }
  const float lam = sqrtf(s1[0]) / sqrtf(s2[0]);
  for (int c = t; c < d; c += 256)
    U[row * d + c] = F[row * d + c] + lam * Mm[row * d + c];
}

// ---------------------------------------------------------------------------
// Row-wise L2 normalize in place (matches x / max(||x||, 1e-12))
// ---------------------------------------------------------------------------
__global__ __launch_bounds__(256) void k_l2norm(float* __restrict__ E, int d)
{
  __shared__ float s1[256];
  const int row = blockIdx.x, t = threadIdx.x;
  float n = 0.f;
  for (int c = t; c < d; c += 256) { const float e = E[row * d + c]; n += e * e; }
  s1[t] = n; __syncthreads();
  for (int s = 128; s > 0; s >>= 1) { if (t < s) s1[t] += s1[t + s]; __syncthreads(); }
  const float inv = 1.f / fmaxf(sqrtf(s1[0]), 1e-12f);
  for (int c = t; c < d; c += 256) E[row * d + c] *= inv;
}

// ---------------------------------------------------------------------------
// Persistent ADMM QP solver + voting softmax. n1=n2=64, n=4096.
// Thread t owns cells (a,b) = (t>>6 + 16k, t&63), k=0..3 (all state in VGPRs).
// K*v uses only row/col/total sums of v and P.*v -> O(n) matvec in LDS.
// z2 == 1 identically (l==u==1 on constraint rows).
// ---------------------------------------------------------------------------
__global__ __launch_bounds__(1024) void k_admm_voting(
    const float* __restrict__ Pg, float* __restrict__ out)
{
  __shared__ float Ps[4096];     // P matrix (n2 x n1)
  __shared__ float Vb[4096];     // matvec staging / final scores
  __shared__ float pd[1024];     // dot-product tree
  __shared__ float Rr[64], Cc[64], TRr[64], TCc[64];
  __shared__ float y2r[64], y2c[64];
  __shared__ float sS[1], sT[1];

  const int t  = (int)threadIdx.x;
  const int b  = t & 63;
  const int a0 = t >> 6;

  int   a[4];
  float x[4], z1[4], y1[4], r[4], pcg[4], ap[4], rhs[4], pm[4];

#pragma unroll
  for (int k = 0; k < 4; ++k) {
    a[k] = a0 + (k << 4);
    const int idx = (a[k] << 6) + b;
    const float p = Pg[idx];
    Ps[idx] = p; pm[k] = p;
    x[k] = 0.f; z1[k] = 0.f; y1[k] = 0.f;
  }
  if (t < 64) { y2r[t] = 0.f; y2c[t] = 0.f; }
  __syncthreads();

  // Deterministic row/col/scalar sums of v and P.*v
  auto computeSums = [&](const float (&v)[4]) {
    __syncthreads();                         // prior consumers of Rr/Cc done
#pragma unroll
    for (int k = 0; k < 4; ++k) Vb[(a[k] << 6) + b] = v[k];
    __syncthreads();
    if (t < 64) {
      const float* vr = &Vb[t << 6];
      const float* pr = &Ps[t << 6];
      float rs = 0.f, trs = 0.f;
      for (int j = 0; j < 64; ++j) { rs += vr[j]; trs += pr[j] * vr[j]; }
      Rr[t] = rs; TRr[t] = trs;
    } else if (t < 128) {
      const int bb = t - 64;
      float cs = 0.f, tcs = 0.f;
      for (int i = 0; i < 64; ++i) {
        const float vv = Vb[(i << 6) + bb];
        cs += vv; tcs += Ps[(i << 6) + bb] * vv;
      }
      Cc[bb] = cs; TCc[bb] = tcs;
    }
    __syncthreads();
    if (t == 0) {
      float s = 0.f, tt = 0.f;
      for (int i = 0; i < 64; ++i) { s += Rr[i]; tt += TRr[i]; }
      sS[0] = s; sT[0] = tt;
    }
    __syncthreads();
  };

  // o = K*v via the structured operator
  auto applyK = [&](const float (&v)[4], float (&o)[4]) {
    computeSums(v);
    const float S = sS[0], T = sT[0];
#pragma unroll
    for (int k = 0; k < 4; ++k) {
      const float ra = Rr[a[k]], cb = Cc[b];
      const float m4v = 0.5f * (pm[k] * (S - ra - cb + v[k]) +
                                (T - TRr[a[k]] - TCc[b] + pm[k] * v[k]));
      o[k] = KDIAG * v[k] + RHO_F * (ra + cb) - m4v;
    }
  };

  // Deterministic block-wide dot product
  auto blockDot = [&](float partial) -> float {
    __syncthreads();
    pd[t] = partial;
    __syncthreads();
    for (int s = 512; s > 0; s >>= 1) {
      if (t < s) pd[t] += pd[t + s];
      __syncthreads();
    }
    const float res = pd[0];
    __syncthreads();
    return res;
  };

  for (int it = 0; it < ADMM_ITERS; ++it) {
    // rhs = sigma*x + P + (rho*z1 - y1) + A^T(rho*1 - y2)
#pragma unroll
    for (int k = 0; k < 4; ++k)
      rhs[k] = SIGMA_F * x[k] + pm[k] + RHO_F * z1[k] - y1[k]
             + (2.f * RHO_F - y2c[b] - y2r[a[k]]);

    // CG solve K x = rhs, warm-started from previous x
    applyK(x, ap);
    float part = 0.f;
#pragma unroll
    for (int k = 0; k < 4; ++k) { r[k] = rhs[k] - ap[k]; pcg[k] = r[k]; part += r[k] * r[k]; }
    float rs = blockDot(part);

    for (int cg = 0; cg < CG_ITERS; ++cg) {
      applyK(pcg, ap);
      part = 0.f;
#pragma unroll
      for (int k = 0; k < 4; ++k) part += pcg[k] * ap[k];
      const float pap   = blockDot(part);
      const float alpha = rs / fmaxf(pap, 1e-30f);
      part = 0.f;
#pragma unroll
      for (int k = 0; k < 4; ++k) {
        x[k] += alpha * pcg[k];
        r[k] -= alpha * ap[k];
        part += r[k] * r[k];
      }
      const float rs2  = blockDot(part);
      const float beta = rs2 / fmaxf(rs, 1e-30f);
      rs = rs2;
#pragma unroll
      for (int k = 0; k < 4; ++k) pcg[k] = r[k] + beta * pcg[k];
    }

    // z/y updates (z2 == 1 identically)
    computeSums(x);                          // Rr/Cc = row/col sums of new x
#pragma unroll
    for (int k = 0; k < 4; ++k) {
      const float c = x[k] + y1[k] * (1.f / RHO_F);
      z1[k] = fminf(fmaxf(c, 0.f), 1e6f);
      y1[k] += RHO_F * (x[k] - z1[k]);
    }
    if (t < 64) { y2c[t] += RHO_F * (Cc[t] - 1.f); y2r[t] += RHO_F * (Rr[t] - 1.f); }
    __syncthreads();
  }

  // s = clip(x,0,1) reshaped (n2,n1)^T, then softmax(ALPHA*s, axis=-1)
#pragma unroll
  for (int k = 0; k < 4; ++k)
    Vb[(a[k] << 6) + b] = fminf(fmaxf(x[k], 0.f), 1.f);
  __syncthreads();
  if (t < 64) {
    float mx = -1e30f;
    for (int j = 0; j < 64; ++j) mx = fmaxf(mx, Vb[(j << 6) + t]);
    float se = 0.f;
    for (int j = 0; j < 64; ++j) se += expf(ALPHA_F * (Vb[(j << 6) + t] - mx));
    const float inv = 1.f / se;
    for (int j = 0; j < 64; ++j)
      out[t * 64 + j] = expf(ALPHA_F * (Vb[(j << 6) + t] - mx)) * inv;
  }
}

// ---------------------------------------------------------------------------
extern "C" void kernel_launch(void* const* d_in, const int* in_sizes, int n_in,
                              void* d_out, int out_size, void* d_ws, size_t ws_size,
                              hipStream_t stream) {
  (void)in_sizes; (void)n_in; (void)out_size; (void)ws_size;
  const float* ft   = (const float*)d_in[0];   // feat_tra (64,512)
  const float* fd   = (const float*)d_in[1];   // feat_det (64,512)
  const float* iou  = (const float*)d_in[2];   // iou_mat  (64,64)
  const float* W    = (const float*)d_in[3];   // W        (512,512)
  const float* bias = (const float*)d_in[4];   // b        (512)

  float* ws  = (float*)d_ws;
  float* Mp0 = ws;             // 64*64
  float* m1  = Mp0 + 4096;     // 64*512 (reused as e1)
  float* m2  = m1 + 32768;     // 64*512 (reused as e2)
  float* u1  = m2 + 32768;     // 64*512
  float* u2  = u1 + 32768;     // 64*512
  float* Pm  = u2 + 32768;     // 64*64

  // Mp0 = feat_tra @ feat_det^T + iou        (NT, M=N=64, K=512)
  k_wmma_gemm<0, 1, true, false><<<2, 256, 0, stream>>>(
      ft, fd, iou, nullptr, Mp0, 64, 64, 512, 512, 512, 64);
  // m1 = Mp0 @ feat_det                      (NN, M=64, N=512, K=64)
  k_wmma_gemm<0, 0, false, false><<<16, 256, 0, stream>>>(
      Mp0, fd, nullptr, nullptr, m1, 64, 512, 64, 64, 512, 512);
  // m2 = Mp0^T @ feat_tra                    (TN, M=64, N=512, K=64)
  k_wmma_gemm<1, 0, false, false><<<16, 256, 0, stream>>>(
      Mp0, ft, nullptr, nullptr, m2, 64, 512, 64, 64, 512, 512);
  // u = f + (||f||/||m||) * m
  k_fuse_lam<<<64, 256, 0, stream>>>(ft, m1, u1, 512);
  k_fuse_lam<<<64, 256, 0, stream>>>(fd, m2, u2, 512);
  // e = relu(u @ W^T + b)                    (NT, M=64, N=512, K=512)
  float* e1 = m1; float* e2 = m2;
  k_wmma_gemm<0, 1, false, true><<<16, 256, 0, stream>>>(
      u1, W, nullptr, bias, e1, 64, 512, 512, 512, 512, 512);
  k_wmma_gemm<0, 1, false, true><<<16, 256, 0, stream>>>(
      u2, W, nullptr, bias, e2, 64, 512, 512, 512, 512, 512);
  // row-wise L2 normalize
  k_l2norm<<<64, 256, 0, stream>>>(e1, 512);
  k_l2norm<<<64, 256, 0, stream>>>(e2, 512);
  // P = Mp^T = e2 @ e1^T                     (NT, M=N=64, K=512)
  k_wmma_gemm<0, 1, false, false><<<2, 256, 0, stream>>>(
      e2, e1, nullptr, nullptr, Pm, 64, 64, 512, 512, 512, 64);
  // Persistent ADMM + voting softmax -> out (1,64,64)
  k_admm_voting<<<1, 1024, 0, stream>>>(Pm, (float*)d_out);
}